// UnifiedPromptLayer_17592186044974
// MI455X (gfx1250) — compile-verified
//
#include <hip/hip_runtime.h>

#define N_NODES   50000
#define N_EDGES   800000
#define IN_FEAT   128
#define PROMPT_DIM 32
#define EF_DIM     32
#define HP_DIM     32
#define OUT_DIM   224   // 128 (agg nf) + 32 (deg*prompt) + 32 (agg ef) + 32 (deg*hp)

typedef __attribute__((ext_vector_type(2))) float v2f;
typedef __attribute__((ext_vector_type(8))) float v8f;

// Guaranteed hardware f32 atomic-add at L2 (no CAS fallback).
__device__ __forceinline__ void gatomic_add_f32(float* p, float v) {
    asm volatile("global_atomic_add_f32 %0, %1, off" :: "v"(p), "v"(v) : "memory");
}

// ---------------------------------------------------------------------------
// Kernel 1: zero the output accumulator and the degree scratch.
// ---------------------------------------------------------------------------
__global__ void zero_kernel(float* __restrict__ out, float* __restrict__ deg) {
    const size_t stride = (size_t)gridDim.x * blockDim.x;
    size_t i = (size_t)blockIdx.x * blockDim.x + threadIdx.x;
    const size_t total4 = (size_t)N_NODES * OUT_DIM / 4;   // 2.8M float4
    float4 z = make_float4(0.f, 0.f, 0.f, 0.f);
    for (size_t k = i; k < total4; k += stride)
        ((float4*)out)[k] = z;
    for (size_t k = i; k < (size_t)N_NODES; k += stride)
        deg[k] = 0.0f;
}

// ---------------------------------------------------------------------------
// Kernel 2: one wave32 per edge. Gather the 128-f32 node row (float4/lane,
// global_load_b128), scatter-accumulate via hardware f32 atomics (resolved in
// L2: out + node table are L2-resident on MI455X's 192 MB L2). Also
// accumulates edge_features (1 col/lane) and in-degree.
// ---------------------------------------------------------------------------
__global__ void edge_kernel(const float* __restrict__ nf,
                            const float* __restrict__ ef,
                            const int*   __restrict__ src,
                            const int*   __restrict__ dst,
                            float* __restrict__ out,
                            float* __restrict__ deg) {
    const int lane = threadIdx.x & 31;
    const int wave = threadIdx.x >> 5;
    const int e    = blockIdx.x * 8 + wave;          // grid sized exactly
    if (e >= N_EDGES) return;

    const int s = src[e];
    const int d = dst[e];

    // Prefetch next block's edge-feature row (gfx1250 global_prefetch_b8).
    const int pe = (e + 8 < N_EDGES) ? (e + 8) : e;
    __builtin_prefetch(&ef[(size_t)pe * EF_DIM + lane], 0, 1);

    // 128-wide node-feature row: float4 per lane.
    const float4 x = ((const float4*)(nf + (size_t)s * IN_FEAT))[lane];
    float* orow = out + (size_t)d * OUT_DIM;
    float* onf  = orow + lane * 4;
    gatomic_add_f32(onf + 0, x.x);
    gatomic_add_f32(onf + 1, x.y);
    gatomic_add_f32(onf + 2, x.z);
    gatomic_add_f32(onf + 3, x.w);

    // 32-wide edge-feature row: one column per lane -> cols [160,192).
    const float fe = ef[(size_t)e * EF_DIM + lane];
    gatomic_add_f32(orow + 160 + lane, fe);

    // in-degree
    if (lane == 0) gatomic_add_f32(deg + d, 1.0f);
}

// ---------------------------------------------------------------------------
// Kernel 3: out[:,128:160) = deg ⊗ prompt, out[:,192:224) = deg ⊗ hp.
// Rank-1 outer products done with V_WMMA_F32_16X16X4_F32: one wave handles a
// 16-node tile; A has deg in its K=0 column (VGPR0, lanes 0-15), B has the
// 16-wide prompt chunk in its K=0 row; K=1..3 are zero. D layout: VGPR r holds
// M = r + 8*(lane>=16), N = lane&15.  No divergent branches -> EXEC all ones.
// ---------------------------------------------------------------------------
__global__ void prompt_kernel(const float* __restrict__ prompt,
                              const float* __restrict__ hp,
                              const float* __restrict__ deg,
                              float* __restrict__ out) {
    const int lane = threadIdx.x;              // blockDim.x == 32
    const int m0   = blockIdx.x * 16;          // 50000/16 = 3125 blocks exact
    const int l15  = lane & 15;
    const bool lo  = (lane < 16);

    // A: 16x4, K=0 column = deg tile, everything else zero.
    const float dv = deg[m0 + l15];            // all lanes load a valid addr
    v2f a; a.x = lo ? dv : 0.0f; a.y = 0.0f;   // v_cndmask, no EXEC change

    // B operands: 4x16, row K=0 = 16-wide chunk of prompt / hetero_prompt.
    const float p0 = prompt[l15], p1 = prompt[16 + l15];
    const float h0 = hp[l15],     h1 = hp[16 + l15];
    v2f b0; b0.x = lo ? p0 : 0.0f; b0.y = 0.0f;
    v2f b1; b1.x = lo ? p1 : 0.0f; b1.y = 0.0f;
    v2f b2; b2.x = lo ? h0 : 0.0f; b2.y = 0.0f;
    v2f b3; b3.x = lo ? h1 : 0.0f; b3.y = 0.0f;

    v8f c = {0.f, 0.f, 0.f, 0.f, 0.f, 0.f, 0.f, 0.f};
    // (neg_a, A, neg_b, B, c_mod, C, reuse_a, reuse_b)
    v8f d0 = __builtin_amdgcn_wmma_f32_16x16x4_f32(false, a, false, b0, (short)0, c, false, false);
    v8f d1 = __builtin_amdgcn_wmma_f32_16x16x4_f32(false, a, false, b1, (short)0, c, false, false);
    v8f d2 = __builtin_amdgcn_wmma_f32_16x16x4_f32(false, a, false, b2, (short)0, c, false, false);
    v8f d3 = __builtin_amdgcn_wmma_f32_16x16x4_f32(false, a, false, b3, (short)0, c, false, false);

    const int mofs = (lane >> 4) * 8;          // lanes 16-31 hold M = 8..15
#pragma unroll
    for (int r = 0; r < 8; ++r) {
        const size_t row = (size_t)(m0 + mofs + r) * OUT_DIM;
        out[row + 128 + l15]      = d0[r];     // deg * prompt[0:16]
        out[row + 144 + l15]      = d1[r];     // deg * prompt[16:32]
        out[row + 192 + l15]      = d2[r];     // deg * hp[0:16]
        out[row + 208 + l15]      = d3[r];     // deg * hp[16:32]
    }
}

extern "C" void kernel_launch(void* const* d_in, const int* in_sizes, int n_in,
                              void* d_out, int out_size, void* d_ws, size_t ws_size,
                              hipStream_t stream) {
    (void)in_sizes; (void)n_in; (void)out_size; (void)ws_size;

    const float* nf     = (const float*)d_in[0];  // [50000,128]
    const float* ef     = (const float*)d_in[1];  // [800000,32]
    const float* prompt = (const float*)d_in[2];  // [1,32]
    const float* hp     = (const float*)d_in[3];  // [1,32]
    const int*   src    = (const int*)d_in[4];    // [800000]
    const int*   dst    = (const int*)d_in[5];    // [800000]

    float* out = (float*)d_out;                   // [50000,224]
    float* deg = (float*)d_ws;                    // [50000] scratch

    zero_kernel<<<2048, 256, 0, stream>>>(out, deg);
    edge_kernel<<<N_EDGES / 8, 256, 0, stream>>>(nf, ef, src, dst, out, deg);
    prompt_kernel<<<N_NODES / 16, 32, 0, stream>>>(prompt, hp, deg, out);
}